// HGCNLayer_18133351924025
// MI455X (gfx1250) — compile-verified
//
#include <hip/hip_runtime.h>
#include <hip/hip_bf16.h>

typedef __attribute__((ext_vector_type(2))) float v2f;
typedef __attribute__((ext_vector_type(8))) float v8f;

#define DEG   32
#define M_SUB 33          // 1 target + 32 neighbors
#define FEAT  256
#define TM    16          // batch rows per GEMM tile
#define LDSPAD 260        // 256 + 4 pad -> bank = (4m+k) % 64, conflict-free

// ---------------------------------------------------------------------------
// Phase 1: per-batch-node subgraph construction (one wave32 per node).
// Lane k maps 1:1 onto neighbor slot k (DEGREE == 32 == wave32).
// ---------------------------------------------------------------------------
__global__ __launch_bounds__(32)
void hgcn_graph_kernel(const int* __restrict__ adj,         // [N, 32]
                       const int* __restrict__ batch_nodes, // [B]
                       int*   __restrict__ s_out,           // [B, 33]
                       float* __restrict__ w_out,           // [B, 33]
                       float* __restrict__ c0_out)          // [B]
{
    const int b    = blockIdx.x;
    const int lane = threadIdx.x;            // 0..31

    __shared__ int                s[M_SUB];
    __shared__ unsigned long long Arow[M_SUB];
    __shared__ float              c[M_SUB];

    const int v = batch_nodes[b];
    if (lane == 0) s[0] = v;
    s[lane + 1] = adj[(long long)v * DEG + lane];
    __syncthreads();

    // Dedup: valid[j] = no i<j with s[i]==s[j].  Lane l handles j = l+1.
    const int sj = s[lane + 1];
    bool dup = false;
    for (int i = 0; i <= lane; ++i) dup |= (s[i] == sj);
    const unsigned long long validmask =
        1ull | (__ballot(!dup) << 1);        // bit j: s[j] is first occurrence

    // Induced adjacency rows as 33-bit masks: A[i][j] = any_k adj[s_i][k]==s_j
    for (int i = 0; i < M_SUB; ++i) {
        const int si = s[i];
        const int nb = adj[(long long)si * DEG + lane];   // lane = neighbor k
        unsigned long long row = 0ull;
        for (int j = 0; j < M_SUB; ++j) {
            if (__ballot(nb == s[j]) != 0ull) row |= (1ull << j);
        }
        if (!((validmask >> i) & 1ull)) row = 0ull;
        row &= validmask;
        if (lane == 0) Arow[i] = row;
    }
    __syncthreads();

    // In-degree per column j, clamp >= 1, c = 1/sqrt(deg)
    for (int j = lane; j < M_SUB; j += 32) {
        int deg = 0;
        for (int i = 0; i < M_SUB; ++i) deg += (int)((Arow[i] >> j) & 1ull);
        if (deg < 1) deg = 1;
        c[j] = 1.0f / sqrtf((float)deg);
    }
    __syncthreads();

    // w_src[i] = c[i] * sum_j A[i][j] * c[j]
    for (int i = lane; i < M_SUB; i += 32) {
        const unsigned long long row = Arow[i];
        float t = 0.0f;
        for (int j = 0; j < M_SUB; ++j)
            if ((row >> j) & 1ull) t += c[j];
        w_out[(long long)b * M_SUB + i] = c[i] * t;
        s_out[(long long)b * M_SUB + i] = s[i];
    }
    if (lane == 0) c0_out[b] = c[0];
}

// ---------------------------------------------------------------------------
// Phase 2: fused gather-mix + fp32 WMMA GEMM + bias/ReLU epilogue.
// 512 threads = 16 waves. Tile: 16 batch rows x 256 feature cols.
// Wave r builds mix row r in LDS; wave w then owns column tile [w*16, w*16+16).
// ---------------------------------------------------------------------------
__global__ __launch_bounds__(512)
void hgcn_gemm_kernel(const float* __restrict__ nf,    // [N, 256]
                      const float* __restrict__ Wm,    // [256, 256]
                      const int*   __restrict__ s_ws,  // [B, 33]
                      const float* __restrict__ w_ws,  // [B, 33]
                      const float* __restrict__ c0_ws, // [B]
                      float*       __restrict__ out,   // [B, 256]
                      int B)
{
    __shared__ float mixs[TM][LDSPAD];
    __shared__ int   s_t[TM * M_SUB];
    __shared__ float w_t[TM * M_SUB];
    __shared__ float c0_t[TM];

    const int tid     = threadIdx.x;
    const int rowbase = blockIdx.x * TM;

    // --- load tile metadata (clamped for safety) ---
    const long long maxMeta = (long long)B * M_SUB - 1;
    for (int t = tid; t < TM * M_SUB; t += 512) {
        long long gi = (long long)rowbase * M_SUB + t;
        if (gi > maxMeta) gi = maxMeta;
        s_t[t] = s_ws[gi];
        w_t[t] = w_ws[gi];
    }
    if (tid < TM) {
        int r = rowbase + tid; if (r > B - 1) r = B - 1;
        c0_t[tid] = c0_ws[r];
    }
    __syncthreads();

    // --- build mix rows: wave r handles batch row r (w is wave-uniform) ---
    {
        const int r    = tid >> 5;
        const int lane = tid & 31;
        float4 a0 = make_float4(0.f, 0.f, 0.f, 0.f);
        float4 a1 = make_float4(0.f, 0.f, 0.f, 0.f);
        for (int i = 0; i < M_SUB; ++i) {
            const float w = w_t[r * M_SUB + i];
            if (w != 0.0f) {                 // wave-uniform -> s_cbranch skip
                const float4* hp =
                    (const float4*)(nf + (long long)s_t[r * M_SUB + i] * FEAT);
                const float4 h0 = hp[lane];
                const float4 h1 = hp[lane + 32];
                a0.x = fmaf(w, h0.x, a0.x); a0.y = fmaf(w, h0.y, a0.y);
                a0.z = fmaf(w, h0.z, a0.z); a0.w = fmaf(w, h0.w, a0.w);
                a1.x = fmaf(w, h1.x, a1.x); a1.y = fmaf(w, h1.y, a1.y);
                a1.z = fmaf(w, h1.z, a1.z); a1.w = fmaf(w, h1.w, a1.w);
            }
        }
        float4* mrow = (float4*)(&mixs[r][0]);   // row stride 1040B, 16B aligned
        mrow[lane]      = a0;
        mrow[lane + 32] = a1;
    }
    __syncthreads();

    // --- WMMA: D[16x16] += A[16x4] * B[4x16], K-loop over 256 ---
    const int waveId = tid >> 5;
    const int lane   = tid & 31;
    const int half   = lane >> 4;        // 0: K=0,1  1: K=2,3 (A/B half split)
    const int m      = lane & 15;        // A row / B,C,D column within tile
    const int n      = waveId * 16 + m;  // global feature column

    v8f acc = {};
#pragma unroll 4
    for (int k = 0; k < FEAT; k += 4) {
        const int kk = k + 2 * half;
        v2f a; a.x = mixs[m][kk];           a.y = mixs[m][kk + 1];
        v2f bb; bb.x = Wm[(long long)kk * FEAT + n];
        bb.y = Wm[(long long)(kk + 1) * FEAT + n];
        acc = __builtin_amdgcn_wmma_f32_16x16x4_f32(
            /*neg_a=*/false, a, /*neg_b=*/false, bb,
            /*c_mod=*/(short)0, acc, /*reuse_a=*/false, /*reuse_b=*/false);
    }

    // --- epilogue: + c0 * h_self, ReLU, store ---
    // Full-tile fast path is block-uniform: no per-element EXEC churn, the
    // 8 stores stay straight-line and clause-able. (B % TM == 0 here, so the
    // remainder path is for safety only.)
    if (rowbase + TM <= B) {
#pragma unroll
        for (int r8 = 0; r8 < 8; ++r8) {
            const int mr   = r8 + (half << 3);   // C layout: lanes 16-31 -> M+8
            const int row  = rowbase + mr;
            const int node = s_t[mr * M_SUB];    // s[row][0] == target node
            const float bias = c0_t[mr] * nf[(long long)node * FEAT + n];
            out[(long long)row * FEAT + n] = fmaxf(acc[r8] + bias, 0.0f);
        }
    } else {
#pragma unroll
        for (int r8 = 0; r8 < 8; ++r8) {
            const int mr  = r8 + (half << 3);
            const int row = rowbase + mr;
            if (row < B) {
                const int   node = s_t[mr * M_SUB];
                const float bias = c0_t[mr] * nf[(long long)node * FEAT + n];
                out[(long long)row * FEAT + n] = fmaxf(acc[r8] + bias, 0.0f);
            }
        }
    }
}

// ---------------------------------------------------------------------------
extern "C" void kernel_launch(void* const* d_in, const int* in_sizes, int n_in,
                              void* d_out, int out_size, void* d_ws, size_t ws_size,
                              hipStream_t stream) {
    const float* nf  = (const float*)d_in[0];   // node_features [N,256]
    const float* Wm  = (const float*)d_in[1];   // weight [256,256]
    const int*   adj = (const int*)d_in[2];     // adj [N,32]
    const int*   bn  = (const int*)d_in[3];     // batch_nodes [B]
    float*       out = (float*)d_out;

    const int B = in_sizes[3];                  // 10000

    // workspace layout: s [B*33 int] | w_src [B*33 f32] | c0 [B f32]
    char*  ws    = (char*)d_ws;
    int*   s_ws  = (int*)ws;
    float* w_ws  = (float*)(ws + (size_t)B * M_SUB * sizeof(int));
    float* c0_ws = (float*)(ws + (size_t)B * M_SUB * (sizeof(int) + sizeof(float)));

    hgcn_graph_kernel<<<B, 32, 0, stream>>>(adj, bn, s_ws, w_ws, c0_ws);

    const int tiles = (B + TM - 1) / TM;        // 625
    hgcn_gemm_kernel<<<tiles, 512, 0, stream>>>(nf, Wm, s_ws, w_ws, c0_ws, out, B);
}